// QuantumFeatureMap_70050916598221
// MI455X (gfx1250) — compile-verified
//
#include <hip/hip_runtime.h>
#include <hip/hip_bf16.h>
#include <math.h>

// QuantumFeatureMap, analytically collapsed (CNOT cascade == Gray-code permutation,
// product state => every expectation factorizes per qubit):
//   out[b, q]     = prod_{q'=0..q}   cos(0.5*tanh(x[b,q']))          (q = 0..7)
//   out[b, (i,j)] = prod_{q'=i+1..j} cos(0.5*tanh(x[b,q']))          (28 pairs)
// Computed exactly in log2 domain as exp2( log2-g @ selection^T ) with
// V_WMMA_F32_16X16X4_F32 on the (B x 8) x (8 x 36) GEMM (K=8 => 2 chained WMMAs,
// N=36 padded to 48 => 3 tiles). Memory-bound: ~23 MB traffic -> ~1 us @ 23.3 TB/s.

typedef __attribute__((ext_vector_type(2)))  float    v2f;
typedef __attribute__((ext_vector_type(8)))  float    v8f;

#define NQ 8
#define NF 36

__device__ __forceinline__ float lgterm(float x) {
    // log2(cos(0.5*tanh(x))); cos argument in (-0.5, 0.5) => cos in (0.877, 1], log2 safe.
    return __log2f(__cosf(0.5f * tanhf(x)));
}

// Feature n -> inclusive qubit range [lo, hi] whose g's are multiplied.
__device__ __forceinline__ void feat_range(int n, int& lo, int& hi) {
    if (n < NQ) { lo = 0; hi = n; }
    else if (n < NF) {
        int p = n - NQ, i = 0;
        while (p >= (NQ - 1) - i) { p -= (NQ - 1) - i; ++i; }   // pair (i, j)
        lo = i + 1; hi = i + 1 + p;
    } else { lo = 1; hi = 0; }  // padded column: empty product
}

__device__ __forceinline__ float bsel(int k, int lo, int hi) {
    return (k >= lo && k <= hi) ? 1.0f : 0.0f;
}

__global__ __launch_bounds__(256)
void qfm_wmma_kernel(const float* __restrict__ x, float* __restrict__ out, int nrows) {
    const int lane  = threadIdx.x & 31;
    const int wave  = threadIdx.x >> 5;
    const int tile  = blockIdx.x * (blockDim.x >> 5) + wave;     // 16 rows per wave
    const int ntile = (nrows + 15) >> 4;
    if (tile >= ntile) return;                                   // whole-wave exit; EXEC stays full

    const int row0 = tile << 4;
    const int m    = lane & 15;        // batch row within tile / output column within N-tile
    const int hi16 = lane >> 4;        // lane half
    const int c0   = hi16 << 1;        // A-matrix K base per WMMA f32 16x16x4 layout: 0 or 2
    const bool fullTile = (row0 + 16) <= nrows;

    int rowm = row0 + m;
    int rowc = rowm < nrows ? rowm : (nrows - 1);                // clamp tail loads
    const float2* xr = (const float2*)(x + (size_t)rowc * NQ);
    float2 u = xr[hi16];          // cols c0,   c0+1   (each x element loaded exactly once/wave)
    float2 w = xr[hi16 + 2];      // cols c0+4, c0+5

    // A operands land directly in WMMA layout: lanes 0-15 -> K=0,1 / 4,5 ; lanes 16-31 -> K=2,3 / 6,7
    v2f alo = { lgterm(u.x), lgterm(u.y) };
    v2f ahi = { lgterm(w.x), lgterm(w.y) };

    const int rbase = row0 + (hi16 << 3);          // C/D layout: VGPR v -> M = v + 8*hi16

    #pragma unroll
    for (int t = 0; t < 3; ++t) {
        const int col = 16 * t + m;
        int lo, hi; feat_range(col, lo, hi);
        v2f blo = { bsel(c0,     lo, hi), bsel(c0 + 1, lo, hi) };
        v2f bhi = { bsel(c0 + 4, lo, hi), bsel(c0 + 5, lo, hi) };

        v8f acc = {};
        acc = __builtin_amdgcn_wmma_f32_16x16x4_f32(false, alo, false, blo, (short)0, acc, false, false);
        acc = __builtin_amdgcn_wmma_f32_16x16x4_f32(false, ahi, false, bhi, (short)0, acc, false, false);

        if (fullTile) {
            if (col < NF) {                                  // uniform in v: one exec toggle per tile
                float* po = out + (size_t)rbase * NF + col;  // 8 stores at immediate offsets v*144B
                #pragma unroll
                for (int v = 0; v < 8; ++v)
                    po[(size_t)v * NF] = __builtin_amdgcn_exp2f(acc[v]);
            }
        } else {                                             // tail tile (unused for B=131072)
            #pragma unroll
            for (int v = 0; v < 8; ++v) {
                int row = rbase + v;
                if (row < nrows && col < NF)
                    out[(size_t)row * NF + col] = __builtin_amdgcn_exp2f(acc[v]);
            }
        }
    }
}

extern "C" void kernel_launch(void* const* d_in, const int* in_sizes, int n_in,
                              void* d_out, int out_size, void* d_ws, size_t ws_size,
                              hipStream_t stream) {
    (void)n_in; (void)out_size; (void)d_ws; (void)ws_size;
    const float* x   = (const float*)d_in[0];
    float*       out = (float*)d_out;
    const int nrows  = in_sizes[0] / NQ;          // 131072
    const int ntiles = (nrows + 15) / 16;
    const int wavesPerBlock = 8;                  // 256 threads = 8 wave32
    dim3 block(256);
    dim3 grid((ntiles + wavesPerBlock - 1) / wavesPerBlock);
    qfm_wmma_kernel<<<grid, block, 0, stream>>>(x, out, nrows);
}